// GATv2Convolution_46411416601106
// MI455X (gfx1250) — compile-verified
//
#include <hip/hip_runtime.h>
#include <hip/hip_bf16.h>

// GATv2-style attention conv, decomposed:
//   u = h @ w[:C], v = h @ w[C:]            (WMMA f32 GEMM, N x C)
//   e_edge = LeakyReLU(u[src]+v[dst]) . a   (wave-per-edge, L2-resident)
//   denom[src] += exp(e)                    (f32 atomic)
//   out[dst]  += h[src] * exp(e)/denom[src] (f32 atomics, 128/edge)

#define C 128
#define LEAKY 0.2f

typedef __attribute__((ext_vector_type(2))) float v2f;
typedef __attribute__((ext_vector_type(8))) float v8f;

// ---------------------------------------------------------------- zero init
__global__ __launch_bounds__(256) void gat_zero(float* __restrict__ out, int n_out,
                                                float* __restrict__ denom, int n_den) {
    int i = blockIdx.x * 256 + threadIdx.x;
    if (i < n_out) out[i] = 0.0f;
    if (i < n_den) denom[i] = 0.0f;
}

// ------------------------------------------------- node projections u,v (WMMA)
// One block = 256 threads = 8 waves, handles 16 rows of h.
// Wave w owns output column tile [16w, 16w+16). Each wave accumulates BOTH the
// u-tile and the v-tile (A fragment shared), 2 wmma per k-step, K=128 -> 64 wmma.
__global__ __launch_bounds__(256) void gat_gemm_uv(const float* __restrict__ h,
                                                   const float* __restrict__ w,
                                                   float* __restrict__ u,
                                                   float* __restrict__ v,
                                                   int n_nodes) {
    __shared__ float lh[16 * C];            // 16 rows of h, 8 KB
    const int row0 = blockIdx.x * 16;
    const int tid  = threadIdx.x;
    const bool full_tile = (row0 + 16) <= n_nodes;

    if (full_tile) {
        for (int i = tid; i < 16 * C; i += 256)
            lh[i] = h[(size_t)row0 * C + i];
    } else {
        for (int i = tid; i < 16 * C; i += 256) {
            int gr = row0 + (i >> 7);
            lh[i] = (gr < n_nodes) ? h[(size_t)gr * C + (i & (C - 1))] : 0.0f;
        }
    }
    __syncthreads();

    const int wave = tid >> 5;              // 0..7 -> column tile
    const int lane = tid & 31;
    const int half = lane >> 4;             // 0: K=k,k+1   1: K=k+2,k+3
    const int l16  = lane & 15;
    const int col  = wave * 16 + l16;       // N coordinate for B/D fragments

    v8f accu = {};
    v8f accv = {};
    for (int k = 0; k < C; k += 4) {
        const int ka = k + half * 2;
        // A 16x4 f32 fragment: lane l16 holds row M=l16, VGPR0/1 = K=ka/ka+1
        v2f afrag;
        afrag.x = lh[l16 * C + ka];
        afrag.y = lh[l16 * C + ka + 1];
        // B 4x16 f32 fragments (N striped across lanes, K across VGPRs/halves)
        v2f bu, bv;
        bu.x = w[(size_t)(ka    ) * C + col];
        bu.y = w[(size_t)(ka + 1) * C + col];
        bv.x = w[(size_t)(C + ka    ) * C + col];
        bv.y = w[(size_t)(C + ka + 1) * C + col];
        accu = __builtin_amdgcn_wmma_f32_16x16x4_f32(false, afrag, false, bu,
                                                     (short)0, accu, false, false);
        accv = __builtin_amdgcn_wmma_f32_16x16x4_f32(false, afrag, false, bv,
                                                     (short)0, accv, false, false);
    }

    // D layout: VGPR r -> M=r (lanes 0-15) / M=r+8 (lanes 16-31), N = col
    const int mbase = row0 + half * 8;
    if (full_tile) {
        #pragma unroll
        for (int r = 0; r < 8; ++r) {
            u[(size_t)(mbase + r) * C + col] = accu[r];
            v[(size_t)(mbase + r) * C + col] = accv[r];
        }
    } else {
        #pragma unroll
        for (int r = 0; r < 8; ++r) {
            if (mbase + r < n_nodes) {
                u[(size_t)(mbase + r) * C + col] = accu[r];
                v[(size_t)(mbase + r) * C + col] = accv[r];
            }
        }
    }
}

__device__ __forceinline__ float leaky(float x) { return x > 0.0f ? x : LEAKY * x; }

// -------------------------------------------- per-edge logits + segment denom
// One wave per edge: 32 lanes x float4 = 128 channels.
__global__ __launch_bounds__(256) void gat_edge_e(const float* __restrict__ u,
                                                  const float* __restrict__ v,
                                                  const float* __restrict__ a,
                                                  const int* __restrict__ src,
                                                  const int* __restrict__ dst,
                                                  float* __restrict__ exp_e,
                                                  float* __restrict__ denom,
                                                  int n_edges) {
    const int wib  = threadIdx.x >> 5;
    const int lane = threadIdx.x & 31;
    const int e = blockIdx.x * 8 + wib;
    if (e >= n_edges) return;

    const int s = src[e];
    const int d = dst[e];
    float4 uu = ((const float4*)(u + (size_t)s * C))[lane];
    float4 vv = ((const float4*)(v + (size_t)d * C))[lane];
    float4 aa = ((const float4*)a)[lane];

    float sum = leaky(uu.x + vv.x) * aa.x;
    sum = fmaf(leaky(uu.y + vv.y), aa.y, sum);
    sum = fmaf(leaky(uu.z + vv.z), aa.z, sum);
    sum = fmaf(leaky(uu.w + vv.w), aa.w, sum);

    #pragma unroll
    for (int off = 16; off > 0; off >>= 1)
        sum += __shfl_xor(sum, off, 32);

    if (lane == 0) {
        float ex = expf(sum);
        exp_e[e] = ex;
        atomicAdd(&denom[s], ex);
    }
}

// ----------------------------------- attention-weighted scatter into out[dst]
__global__ __launch_bounds__(256) void gat_scatter(const float* __restrict__ h,
                                                   const int* __restrict__ src,
                                                   const int* __restrict__ dst,
                                                   const float* __restrict__ exp_e,
                                                   const float* __restrict__ denom,
                                                   float* __restrict__ out,
                                                   int n_edges) {
    const int wib  = threadIdx.x >> 5;
    const int lane = threadIdx.x & 31;
    const int e = blockIdx.x * 8 + wib;
    if (e >= n_edges) return;

    const int s = src[e];
    const int d = dst[e];
    const float att = exp_e[e] / denom[s];

    float4 hv = ((const float4*)(h + (size_t)s * C))[lane];
    float* op = out + (size_t)d * C + lane * 4;
    atomicAdd(op + 0, hv.x * att);
    atomicAdd(op + 1, hv.y * att);
    atomicAdd(op + 2, hv.z * att);
    atomicAdd(op + 3, hv.w * att);
}

extern "C" void kernel_launch(void* const* d_in, const int* in_sizes, int n_in,
                              void* d_out, int out_size, void* d_ws, size_t ws_size,
                              hipStream_t stream) {
    const float* h  = (const float*)d_in[0];
    const float* w  = (const float*)d_in[1];
    const float* a  = (const float*)d_in[2];
    const int* src  = (const int*)d_in[3];
    const int* dst  = (const int*)d_in[4];
    float* out      = (float*)d_out;

    const int n_nodes = in_sizes[0] / C;
    const int n_edges = in_sizes[3];

    // workspace: u[N*C] | v[N*C] | exp_e[E] | denom[N]   (~23 MB)
    float* u     = (float*)d_ws;
    float* v     = u + (size_t)n_nodes * C;
    float* exp_e = v + (size_t)n_nodes * C;
    float* denom = exp_e + n_edges;

    {   // zero accumulators (out + denom) every call
        int nz = n_nodes * C;
        int grid = (nz + 255) / 256;
        gat_zero<<<grid, 256, 0, stream>>>(out, nz, denom, n_nodes);
    }
    {   // node projections via f32 WMMA
        int grid = (n_nodes + 15) / 16;
        gat_gemm_uv<<<grid, 256, 0, stream>>>(h, w, u, v, n_nodes);
    }
    {   // edge logits + softmax denominator
        int grid = (n_edges + 7) / 8;
        gat_edge_e<<<grid, 256, 0, stream>>>(u, v, a, src, dst, exp_e, denom, n_edges);
    }
    {   // weighted scatter
        int grid = (n_edges + 7) / 8;
        gat_scatter<<<grid, 256, 0, stream>>>(h, src, dst, exp_e, denom, out, n_edges);
    }
}